// SplitBruteForceAntisymmetrize_36532991819879
// MI455X (gfx1250) — compile-verified
//
#include <hip/hip_runtime.h>
#include <math.h>

// ---------------------------------------------------------------------------
// Antisymmetrized 2-leaf MLP:  out[b] = log|psi1(b)| + log|psi2(b)|
//   psi(b) = sum_p sign(p) * ( tanh( x_perm(b,p) @ W1 + b1 ) @ W2 + b2 )
//
// Phase 1 (WMMA fp32): per batch element, C[i][j][:] = x[i,:] @ W1[3j:3j+3,:]
//   as C(36x64) = A'(36x20) @ W1(20x64) with V_WMMA_F32_16X16X4_F32.
// Phase 2: lexicographic enumeration of the 720 perms with prefix sums over
//   slots (shared subtree work), hardware tanh, pairwise sign cancellation
//   of the b2 term. One wave32 per batch element; both leaves sequential.
// ---------------------------------------------------------------------------

#define NPART 6
#define ND    18            // N*D
#define HID   64
#define WPB   4             // waves per block
#define BLOCK (WPB * 32)

typedef float v2f __attribute__((ext_vector_type(2)));
typedef float v8f __attribute__((ext_vector_type(8)));

#if __has_builtin(__builtin_amdgcn_tanhf)
#define TANH(x) __builtin_amdgcn_tanhf(x)
#else
#define TANH(x) tanhf(x)
#endif

// remove d-th nibble from ascending nibble list
__device__ __forceinline__ unsigned nib_remove(unsigned lst, int d) {
    const unsigned low = (1u << (4 * d)) - 1u;
    return (lst & low) | ((lst >> (4 * (d + 1))) << (4 * d));
}

// load C[pick][slot][2*lane : 2*lane+2] ; cb = cbuf + 2*lane
__device__ __forceinline__ v2f loadC(const float* __restrict__ cb, unsigned pick, int slot) {
    return *(const v2f*)(cb + pick * (NPART * HID) + slot * HID);
}

__device__ __forceinline__ float leaf_eval(
    const float* __restrict__ x,     // (B, 18)
    const float* __restrict__ W1,    // (18, 64) row-major
    const float* __restrict__ b1,    // (64)
    const float* __restrict__ W2,    // (64)
    float* __restrict__ cbuf,        // per-wave LDS: 36*64 floats
    float* __restrict__ xs,          // per-wave LDS: 20 floats (2 pad zeros)
    int b, int lane)
{
    // ---- stage x[b] into LDS (with zero padding slots 18,19) --------------
    if (lane < 20) xs[lane] = (lane < ND) ? x[b * ND + lane] : 0.0f;
    __syncthreads();

    const int lmod = lane & 15;
    const int hi   = lane >> 4;        // 0 or 1 (K half select)

    // ---- preload W1 fragments (B-matrix, 4x16 tiles, K padded 18->20) -----
    v2f bf[5][4];
#pragma unroll
    for (int kt = 0; kt < 5; ++kt) {
#pragma unroll
        for (int nt = 0; nt < 4; ++nt) {
            const int n = nt * 16 + lmod;
            const int K = kt * 4 + hi * 2;
            v2f f;
            f.x = (K     < ND) ? W1[K * HID + n]       : 0.0f;
            f.y = (K + 1 < ND) ? W1[(K + 1) * HID + n] : 0.0f;
            bf[kt][nt] = f;
        }
    }

    // ---- phase 1: C(36x64) = A'(36x20) @ W1(20x64) via WMMA f32 16x16x4 ---
    // Row r = i*6 + j of A' holds x[i,0:3] at columns 3j..3j+2, zero elsewhere.
#pragma unroll
    for (int mt = 0; mt < 3; ++mt) {
        v8f acc[4] = {};
        const int row   = mt * 16 + lmod;
        const int iq    = row / 6;          // folded magic-mul
        const int j     = row - 6 * iq;
        const int i3    = 3 * iq;
        const bool rowok = (row < 36);
#pragma unroll
        for (int kt = 0; kt < 5; ++kt) {
            // element K values: half0 -> kt*4+e ; half1 -> kt*4+2+e  (e=0,1)
            // all K-derived terms are selects between compile-time constants
            const int jK0 = hi ? ((kt * 4 + 2) / 3) : ((kt * 4 + 0) / 3);
            const int rK0 = hi ? ((kt * 4 + 2) % 3) : ((kt * 4 + 0) % 3);
            const bool ok0 = hi ? ((kt * 4 + 2) < ND) : ((kt * 4 + 0) < ND);
            const int jK1 = hi ? ((kt * 4 + 3) / 3) : ((kt * 4 + 1) / 3);
            const int rK1 = hi ? ((kt * 4 + 3) % 3) : ((kt * 4 + 1) % 3);
            const bool ok1 = hi ? ((kt * 4 + 3) < ND) : ((kt * 4 + 1) < ND);

            const int idx0 = (rowok && ok0 && (jK0 == j)) ? (i3 + rK0) : 19;
            const int idx1 = (rowok && ok1 && (jK1 == j)) ? (i3 + rK1) : 19;
            v2f a;
            a.x = xs[idx0];
            a.y = xs[idx1];
#pragma unroll
            for (int nt = 0; nt < 4; ++nt) {
                acc[nt] = __builtin_amdgcn_wmma_f32_16x16x4_f32(
                    false, a, false, bf[kt][nt], (short)0, acc[nt], false, false);
            }
        }
        // scatter C tile to LDS: lane half h, VGPR r -> M = mt*16 + r + h*8
#pragma unroll
        for (int nt = 0; nt < 4; ++nt) {
#pragma unroll
            for (int r = 0; r < 8; ++r) {
                const int M = mt * 16 + r + hi * 8;
                if (M < 36) cbuf[M * HID + nt * 16 + lmod] = acc[nt][r];
            }
        }
    }
    __syncthreads();

    // ---- phase 2: lexicographic sweep with prefix sums --------------------
    // lane owns hidden pair (2*lane, 2*lane+1)
    const float w0 = W2[2 * lane];
    const float w1 = W2[2 * lane + 1];
    const float* cb = cbuf + 2 * lane;
    v2f sbias;
    sbias.x = b1[2 * lane];
    sbias.y = b1[2 * lane + 1];

    float acc = 0.0f;
    const unsigned lst0 = 0x543210u;
    for (int d0 = 0; d0 < 6; ++d0) {
        const unsigned lst1 = nib_remove(lst0, d0);
        const v2f s0 = sbias + loadC(cb, (unsigned)d0, 0);   // identity list: pick == d0
        for (int d1 = 0; d1 < 5; ++d1) {
            const unsigned q1 = (lst1 >> (4 * d1)) & 0xFu;
            const unsigned lst2 = nib_remove(lst1, d1);
            const v2f s1 = s0 + loadC(cb, q1, 1);
            for (int d2 = 0; d2 < 4; ++d2) {
                const unsigned q2 = (lst2 >> (4 * d2)) & 0xFu;
                const unsigned lst3 = nib_remove(lst2, d2);
                const v2f s2 = s1 + loadC(cb, q2, 2);
                const int par2 = (d0 + d1 + d2) & 1;
#pragma unroll
                for (int d3 = 0; d3 < 3; ++d3) {
                    const unsigned q3 = (lst3 >> (4 * d3)) & 0xFu;
                    const unsigned lst4 = nib_remove(lst3, d3);
                    const v2f s3 = s2 + loadC(cb, q3, 3);
                    const unsigned qa = lst4 & 0xFu;          // qa < qb (ascending)
                    const unsigned qb = (lst4 >> 4) & 0xFu;
                    // perm A: slots 4,5 = (qa,qb)   parity par2^d3
                    // perm B: slots 4,5 = (qb,qa)   parity par2^d3^1
                    const v2f pA = s3 + loadC(cb, qa, 4) + loadC(cb, qb, 5);
                    const v2f pB = s3 + loadC(cb, qb, 4) + loadC(cb, qa, 5);
                    const float fA = fmaf(TANH(pA.x), w0, TANH(pA.y) * w1);
                    const float fB = fmaf(TANH(pB.x), w0, TANH(pB.y) * w1);
                    const float sgn = ((par2 ^ d3) & 1) ? -1.0f : 1.0f;
                    // b2 contribution cancels exactly within each (A,B) pair
                    acc = fmaf(sgn, fA - fB, acc);
                }
            }
        }
    }

    // butterfly all-reduce across the wave (dot product spans all 32 lanes)
#pragma unroll
    for (int m = 16; m >= 1; m >>= 1) acc += __shfl_xor(acc, m, 32);

    __syncthreads();   // cbuf reused by next leaf
    return acc;
}

__global__ __launch_bounds__(BLOCK)
void antisym_kernel(const float* __restrict__ x1, const float* __restrict__ x2,
                    const float* __restrict__ W1a, const float* __restrict__ b1a,
                    const float* __restrict__ W2a,
                    const float* __restrict__ W1b, const float* __restrict__ b1b,
                    const float* __restrict__ W2b,
                    float* __restrict__ out, int batch)
{
    __shared__ __align__(16) float s_cbuf[WPB][36 * HID];
    __shared__ float s_x[WPB][20];

    const int wave = threadIdx.x >> 5;
    const int lane = threadIdx.x & 31;
    const int b = blockIdx.x * WPB + wave;
    const int beff = (b < batch) ? b : (batch - 1);   // keep control flow uniform

    const float psi1 = leaf_eval(x1, W1a, b1a, W2a,
                                 s_cbuf[wave], s_x[wave], beff, lane);
    const float psi2 = leaf_eval(x2, W1b, b1b, W2b,
                                 s_cbuf[wave], s_x[wave], beff, lane);

    if (b < batch && lane == 0)
        out[b] = logf(fabsf(psi1)) + logf(fabsf(psi2));
}

extern "C" void kernel_launch(void* const* d_in, const int* in_sizes, int n_in,
                              void* d_out, int out_size, void* d_ws, size_t ws_size,
                              hipStream_t stream) {
    const float* x1  = (const float*)d_in[0];
    const float* x2  = (const float*)d_in[1];
    const float* W1a = (const float*)d_in[2];
    const float* b1a = (const float*)d_in[3];
    const float* W2a = (const float*)d_in[4];
    // d_in[5] = b2a : cancels exactly in the signed pair sum
    const float* W1b = (const float*)d_in[6];
    const float* b1b = (const float*)d_in[7];
    const float* W2b = (const float*)d_in[8];
    // d_in[9] = b2b : cancels exactly
    float* out = (float*)d_out;

    const int batch = out_size;                       // 2048
    const int blocks = (batch + WPB - 1) / WPB;       // 512
    antisym_kernel<<<blocks, BLOCK, 0, stream>>>(
        x1, x2, W1a, b1a, W2a, W1b, b1b, W2b, out, batch);
}